// RSModel_10763188044347
// MI455X (gfx1250) — compile-verified
//
#include <hip/hip_runtime.h>

typedef __attribute__((ext_vector_type(2))) float v2f;
typedef __attribute__((ext_vector_type(8))) float v8f;

// out[bn, y, x] = vs[y]*ms[x] + vo[y]*mo[x]  (rank-2 outer product per image)
// One block per bn image; wave ty handles tile-row ty; 16x16 tiles via
// V_WMMA_F32_16X16X4_F32 (K=2 live, K=2..3 zero).
__global__ void __launch_bounds__(512)
rank2_outer_wmma_kernel(const float* __restrict__ rel,   // [BN, C]
                        const int* __restrict__ bbox,    // [BN, 8]
                        float* __restrict__ out,         // [BN, W, W]
                        int C, int W, int tiles)         // tiles = W/16
{
    const int bn   = blockIdx.x;
    const int ty   = threadIdx.x >> 5;        // wave id = tile row
    const int lane = threadIdx.x & 31;        // wave32
    const int half = lane >> 4;               // 0: K=0/1 data, 1: K=2/3 (zero)
    const int lr   = lane & 15;

    // ---- block-uniform bbox decode (scalar loads) ----
    const int* bb = bbox + bn * 8;
    const int sx1 = bb[0] >> 1, sy1 = bb[1] >> 1, sx2 = bb[2] >> 1, sy2 = bb[3] >> 1;
    const int ox1 = bb[4] >> 1, oy1 = bb[5] >> 1, ox2 = bb[6] >> 1, oy2 = bb[7] >> 1;
    const int sh = sy2 - sy1, sw = sx2 - sx1;
    const int oh = oy2 - oy1, ow = ox2 - ox1;
    const bool valid = (sh >= 5) && (sw >= 5) && (oh >= 5) && (ow >= 5);

    // ---- A operand: per-row gathered feature values (loop-invariant) ----
    // A is 16x4 f32: VGPR0 = K0 (lanes 0-15) / K2 (lanes 16-31); VGPR1 = K1/K3.
    const int row = ty * 16 + lr;
    float vs = 0.0f, vo = 0.0f;
    if (half == 0 && valid) {
        const float* relbn = rel + (size_t)bn * C;
        if (row >= sy1 && row < sy2) {
            int r = (row - sy1) * C / (sh > 0 ? sh : 1);
            r = r < C - 1 ? r : C - 1;
            vs = relbn[r];
        }
        if (row >= oy1 && row < oy2) {
            int r = (row - oy1) * C / (oh > 0 ? oh : 1);
            r = r < C - 1 ? r : C - 1;
            vo = relbn[r];
        }
    }
    v2f A;
    A.x = vs;   // K=0 (lanes 16-31 hold K=2 -> already 0 via half test)
    A.y = vo;   // K=1 (K=3 -> 0)

    // per-lane output base: lanes 0-15 -> rows M=0..7 of tile, 16-31 -> M=8..15
    float* obase = out + ((size_t)bn * W + (size_t)(ty * 16 + half * 8)) * W + lr;

    for (int tx = 0; tx < tiles; ++tx) {
        // ---- B operand: 0/1 column masks. VGPR0 = K0/K2, VGPR1 = K1/K3. ----
        const int col = tx * 16 + lr;
        v2f Bv;
        Bv.x = (half == 0 && col >= sx1 && col < sx2) ? 1.0f : 0.0f;
        Bv.y = (half == 0 && col >= ox1 && col < ox2) ? 1.0f : 0.0f;

        v8f c = {};
        // D(16x16) = A(16x4) x B(4x16): vs[y]*ms[x] + vo[y]*mo[x]
        v8f d = __builtin_amdgcn_wmma_f32_16x16x4_f32(
            /*neg_a=*/false, A, /*neg_b=*/false, Bv,
            /*c_mod=*/(short)0, c, /*reuse_a=*/false, /*reuse_b=*/false);

        // D layout: VGPR r, lanes 0-15 -> (M=r, N=lane), lanes 16-31 -> (M=r+8).
        float* op = obase + tx * 16;
        #pragma unroll
        for (int r = 0; r < 8; ++r)
            op[(size_t)r * W] = d[r];
    }
}

extern "C" void kernel_launch(void* const* d_in, const int* in_sizes, int n_in,
                              void* d_out, int out_size, void* d_ws, size_t ws_size,
                              hipStream_t stream) {
    const float* rel  = (const float*)d_in[0];   // [B,N,C] f32
    const int*   bbox = (const int*)d_in[1];     // [B,N,8] i32
    // derive shapes: BN from bbox, C from features, W from output
    const int BN = in_sizes[1] / 8;              // 2048
    const int C  = in_sizes[0] / BN;             // 256
    const int WW = out_size / BN;                // W*W
    int W = 1;
    while (W * W < WW) ++W;                      // W = 128
    const int tiles = W / 16;                    // 8
    dim3 block(tiles * 32);                      // one wave per tile-row (256 thr)
    dim3 grid(BN);
    rank2_outer_wmma_kernel<<<grid, block, 0, stream>>>(
        rel, bbox, (float*)d_out, C, W, tiles);
}